// FixedRadiusSearch_66786741453249
// MI455X (gfx1250) — compile-verified
//
#include <hip/hip_runtime.h>

typedef __attribute__((ext_vector_type(2))) float v2f;
typedef __attribute__((ext_vector_type(8))) float v8f;

#define MAXN 64
#define CAP  128   // per-query candidate cap; expected max in-radius ~30

// One wave per 16-query tile. Pairwise d^2 via V_WMMA_F32_16X16X4_F32:
// A = 16x4 query coords (xyz + 0 pad), B = 4x16 point coords, D = q.p dots.
__launch_bounds__(32)
__global__ void frs_main(const float* __restrict__ points,
                         const float* __restrict__ queries,
                         const float* __restrict__ radius,
                         float* __restrict__ out_idx,
                         float* __restrict__ out_dist,
                         int*   __restrict__ counts,
                         int npoints) {
  __shared__ float s_q2[16];
  __shared__ int   s_cnt[16];
  __shared__ float s_cd[16][CAP];
  __shared__ int   s_ci[16][CAP];

  const int lane = threadIdx.x;   // 0..31
  const int half = lane >> 4;     // 0 | 1
  const int sub  = lane & 15;
  const int q    = blockIdx.x * 16 + sub;

  const float r2 = radius[0] * radius[0];

  // ---- query tile -> A matrix (16x4 f32). lanes 0-15: K=0,1; lanes 16-31: K=2,3
  float qx = queries[3 * q + 0];
  float qy = queries[3 * q + 1];
  float qz = queries[3 * q + 2];
  float q2 = qx * qx + qy * qy + qz * qz;
  if (half == 0) { s_q2[sub] = q2; s_cnt[sub] = 0; }
  v2f a;
  a.x = half ? qz : qx;
  a.y = half ? 0.0f : qy;
  __syncthreads();

  float q2v[8];
  #pragma unroll
  for (int r = 0; r < 8; ++r) q2v[r] = s_q2[r + 8 * half];  // M = r + 8*half

  const int ntiles = npoints >> 4;
  for (int t = 0; t < ntiles; ++t) {
    const int p = t * 16 + sub;
    float px = points[3 * p + 0];
    float py = points[3 * p + 1];
    float pz = points[3 * p + 2];
    float p2 = px * px + py * py + pz * pz;
    v2f b;
    b.x = half ? pz : px;
    b.y = half ? 0.0f : py;

    v8f c = {};
    c = __builtin_amdgcn_wmma_f32_16x16x4_f32(
        /*neg_a=*/false, a, /*neg_b=*/false, b,
        /*c_mod=*/(short)0, c, /*reuse_a=*/false, /*reuse_b=*/false);

    #pragma unroll
    for (int r = 0; r < 8; ++r) {
      // D[m][n] = q[m].p[n]; this lane: m = r + 8*half, n = sub
      float d2 = fmaxf(q2v[r] + p2 - 2.0f * c[r], 0.0f);
      if (d2 <= r2) {
        int m = r + 8 * half;
        int slot = atomicAdd(&s_cnt[m], 1);     // ds_add_rtn_u32
        if (slot < CAP) { s_cd[m][slot] = d2; s_ci[m][slot] = p; }
      }
    }
  }
  __syncthreads();

  // ---- per-query sort + emit (lanes 0..15, one query each) ----
  if (half == 0) {
    const int qq = sub;
    int cnt = s_cnt[qq];
    counts[q] = cnt;                    // true count for row_splits
    int n = cnt < CAP ? cnt : CAP;
    // insertion sort by (d2, idx) -> deterministic regardless of atomic order
    for (int i = 1; i < n; ++i) {
      float d = s_cd[qq][i];
      int  ix = s_ci[qq][i];
      int j = i - 1;
      while (j >= 0 &&
             (s_cd[qq][j] > d || (s_cd[qq][j] == d && s_ci[qq][j] > ix))) {
        s_cd[qq][j + 1] = s_cd[qq][j];
        s_ci[qq][j + 1] = s_ci[qq][j];
        --j;
      }
      s_cd[qq][j + 1] = d;
      s_ci[qq][j + 1] = ix;
    }
    int nvalid = cnt < MAXN ? cnt : MAXN;
    #pragma unroll 4
    for (int k = 0; k < MAXN; ++k) {
      bool v = k < nvalid;
      out_idx [q * MAXN + k] = v ? (float)s_ci[qq][k] : -1.0f;
      out_dist[q * MAXN + k] = v ? s_cd[qq][k] : 0.0f;
    }
  }
}

// Exclusive prefix sum of counts -> row_splits (Q+1 values), single block.
__launch_bounds__(1024)
__global__ void frs_scan(const int* __restrict__ counts,
                         float* __restrict__ splits, int Q) {
  __shared__ int buf[1024];
  __shared__ int s_carry;
  const int tid = threadIdx.x;
  if (tid == 0) s_carry = 0;
  __syncthreads();
  for (int base = 0; base < Q; base += 1024) {
    buf[tid] = counts[base + tid];
    __syncthreads();
    for (int off = 1; off < 1024; off <<= 1) {
      int t = (tid >= off) ? buf[tid - off] : 0;
      __syncthreads();
      buf[tid] += t;
      __syncthreads();
    }
    int incl = buf[tid] + s_carry;
    splits[base + tid + 1] = (float)incl;
    __syncthreads();
    if (tid == 1023) s_carry = incl;
    __syncthreads();
  }
  if (tid == 0) splits[0] = 0.0f;
}

extern "C" void kernel_launch(void* const* d_in, const int* in_sizes, int n_in,
                              void* d_out, int out_size, void* d_ws, size_t ws_size,
                              hipStream_t stream) {
  const float* points  = (const float*)d_in[0];
  const float* queries = (const float*)d_in[1];
  const float* radius  = (const float*)d_in[2];
  const int N = in_sizes[0] / 3;   // 16384
  const int Q = in_sizes[1] / 3;   // 16384

  // d_out (float) layout, concatenated in reference return order:
  //   [Q*64] neighbors_index | [Q+1] row_splits | [Q*64] neighbors_distance
  float* out       = (float*)d_out;
  float* out_idx   = out;
  float* out_split = out + (size_t)Q * MAXN;
  float* out_dist  = out_split + (Q + 1);

  int* counts = (int*)d_ws;        // Q ints = 64 KB scratch

  frs_main<<<Q / 16, 32, 0, stream>>>(points, queries, radius,
                                      out_idx, out_dist, counts, N);
  frs_scan<<<1, 1024, 0, stream>>>(counts, out_split, Q);
}